// Voxelization_8993661517909
// MI455X (gfx1250) — compile-verified
//
#include <hip/hip_runtime.h>

// Problem constants (from reference)
#define P_CNT   64
#define N_PTS   512
#define ROWLEN  128          // FEAT(125) + 3
#define VGRID   12
#define MAX_VOX 1728         // 12^3
#define MAX_PTS 4
#define NCOLT   (MAX_VOX / 16)   // 108 column tiles of 16 voxels
#define BIGF    1.0e38f

typedef float v2f __attribute__((ext_vector_type(2)));
typedef float v8f __attribute__((ext_vector_type(8)));

// ---------------------------------------------------------------------------
// Kernel 1: zero-fill the 226MB output (bandwidth floor of this problem).
// ---------------------------------------------------------------------------
__global__ void voxz_zero_kernel(float4* __restrict__ o, long long n4) {
    long long i = (long long)blockIdx.x * blockDim.x + threadIdx.x;
    const long long stride = (long long)gridDim.x * blockDim.x;
    float4 z; z.x = 0.f; z.y = 0.f; z.z = 0.f; z.w = 0.f;
    for (; i < n4; i += stride) o[i] = z;
}

// ---------------------------------------------------------------------------
// Kernel 2: one workgroup (256 threads = 8 waves) per proposal.
//   Phase 1: rotate points into box frame, block-reduce lo/hi bounds.
//   Phase 2: WMMA f32 16x16x4 GEMM tiles: score = |vc|^2 - 2 x.vc (masked),
//            running per-lane argmin + branchless shfl_xor argmin -> sel[n].
//   Phase 3: stable rank via prefix count over sel[] in LDS.
//   Phase 4: coalesced scatter of rotated 128-float rows into out.
// ---------------------------------------------------------------------------
__global__ __launch_bounds__(256) void voxelize_kernel(
        const float* __restrict__ prop,
        const float* __restrict__ xf,
        float*       __restrict__ out) {
    const int p    = blockIdx.x;
    const int tid  = threadIdx.x;
    const int lane = tid & 31;
    const int wv   = tid >> 5;

    __shared__ float sx[N_PTS], sy[N_PTS], sz[N_PTS];
    __shared__ int   sel[N_PTS];
    __shared__ int   rnk[N_PTS];
    __shared__ float bnd[6];          // lox loy loz hix hiy hiz
    __shared__ float wred[8][6];      // per-wave partial min/max

    // ---- proposal parameters (uniform, every thread recomputes) ----
    const float tx = prop[p * 7 + 0];
    const float ty = prop[p * 7 + 1];
    const float tz = prop[p * 7 + 2];
    const float hh = prop[p * 7 + 3] + 1.0f;   // + 2*DELTA
    const float ww = prop[p * 7 + 4] + 1.0f;
    const float ll = prop[p * 7 + 5] + 1.0f;
    const float ry = prop[p * 7 + 6];
    const float cs = cosf(ry);
    const float sn = sinf(ry);

    // ---- Phase 1: rotate points, local bounds ----
    float mnx =  __builtin_inff(), mny =  __builtin_inff(), mnz =  __builtin_inff();
    float mxx = -__builtin_inff(), mxy = -__builtin_inff(), mxz = -__builtin_inff();
    for (int n = tid; n < N_PTS; n += 256) {
        const size_t ib = ((size_t)p * N_PTS + n) * ROWLEN;
        const float gx = xf[ib + 0];
        const float gy = xf[ib + 1];
        const float gz = xf[ib + 2];
        const float dx = gx - tx, dy = gy - ty, dz = gz - tz;
        const float rx = dx * cs - dz * sn;   // (g - t) @ C, col 0
        const float rz = dx * sn + dz * cs;   // col 2
        sx[n] = rx; sy[n] = dy; sz[n] = rz;
        mnx = fminf(mnx, rx); mxx = fmaxf(mxx, rx);
        mny = fminf(mny, dy); mxy = fmaxf(mxy, dy);
        mnz = fminf(mnz, rz); mxz = fmaxf(mxz, rz);
    }
    #pragma unroll
    for (int off = 16; off > 0; off >>= 1) {
        mnx = fminf(mnx, __shfl_xor(mnx, off, 32));
        mny = fminf(mny, __shfl_xor(mny, off, 32));
        mnz = fminf(mnz, __shfl_xor(mnz, off, 32));
        mxx = fmaxf(mxx, __shfl_xor(mxx, off, 32));
        mxy = fmaxf(mxy, __shfl_xor(mxy, off, 32));
        mxz = fmaxf(mxz, __shfl_xor(mxz, off, 32));
    }
    if (lane == 0) {
        wred[wv][0] = mnx; wred[wv][1] = mny; wred[wv][2] = mnz;
        wred[wv][3] = mxx; wred[wv][4] = mxy; wred[wv][5] = mxz;
    }
    __syncthreads();
    if (tid == 0) {
        float a0 = wred[0][0], a1 = wred[0][1], a2 = wred[0][2];
        float b0 = wred[0][3], b1 = wred[0][4], b2 = wred[0][5];
        for (int q = 1; q < 8; ++q) {
            a0 = fminf(a0, wred[q][0]); a1 = fminf(a1, wred[q][1]); a2 = fminf(a2, wred[q][2]);
            b0 = fmaxf(b0, wred[q][3]); b1 = fmaxf(b1, wred[q][4]); b2 = fmaxf(b2, wred[q][5]);
        }
        const float hx = ll / 24.0f, hy = hh / 24.0f, hz = ww / 24.0f; // (l,h,w)/(2V)
        bnd[0] = a0 - hx; bnd[1] = a1 - hy; bnd[2] = a2 - hz;
        bnd[3] = b0 + hx; bnd[4] = b1 + hy; bnd[5] = b2 + hz;
    }
    __syncthreads();

    const float lox = bnd[0], loy = bnd[1], loz = bnd[2];
    const float hix = bnd[3], hiy = bnd[4], hiz = bnd[5];

    // ---- Phase 2: WMMA score tiles + argmin ----
    // A (16x4 f32):  lanes 0-15 hold M=lane { VGPR0:K0=x', VGPR1:K1=y' },
    //                lanes16-31 hold M=lane-16 { VGPR0:K2=z', VGPR1:K3=1 }.
    // B (4x16 f32):  lanes 0-15 N=lane { VGPR0:K0=-2vcx, VGPR1:K1=-2vcy },
    //                lanes16-31 N=lane-16 { VGPR0:K2=-2vcz, VGPR1:K3=q }.
    // D (16x16 f32): reg r, lane L -> row = r + 8*(L>=16), col = L&15.
    const bool hiHalf = (lane >= 16);
    const int  M      = lane & 15;

    for (int rt = wv; rt < N_PTS / 16; rt += 8) {
        const int row = rt * 16 + M;
        v2f A;
        A.x = hiHalf ? sz[row] : sx[row];
        A.y = hiHalf ? 1.0f    : sy[row];

        float best[8]; int bidx[8];
        #pragma unroll
        for (int r = 0; r < 8; ++r) { best[r] = __builtin_inff(); bidx[r] = 0; }

        for (int t = 0; t < NCOLT; ++t) {
            const int m = t * 16 + M;                 // voxel id for this lane's column
            const int vi = m / 144;
            const int vj = (m / 12) % 12;
            const int vk = m % 12;
            const float vcx = (ll * (float)(vi - 6)) / 12.0f + ll / 24.0f;
            const float vcy = -(hh * (float)vj) / 12.0f - hh / 24.0f;
            const float vcz = (ww * (float)(vk - 6)) / 12.0f + ww / 24.0f;
            const bool inb = (vcx > lox) & (vcx < hix) &
                             (vcy > loy) & (vcy < hiy) &
                             (vcz > loz) & (vcz < hiz);
            const float q = inb ? (vcx * vcx + vcy * vcy + vcz * vcz) : BIGF;
            v2f B;
            B.x = hiHalf ? (-2.0f * vcz) : (-2.0f * vcx);
            B.y = hiHalf ? q             : (-2.0f * vcy);

            v8f C = {0.f, 0.f, 0.f, 0.f, 0.f, 0.f, 0.f, 0.f};
            v8f D = __builtin_amdgcn_wmma_f32_16x16x4_f32(
                        false, A, false, B, (short)0, C, false, false);

            #pragma unroll
            for (int r = 0; r < 8; ++r) {
                const float scv = D[r];
                const bool take = scv < best[r];
                best[r] = take ? scv : best[r];
                bidx[r] = take ? m   : bidx[r];
            }
        }

        // cross-lane argmin within each 16-lane half (tie -> lowest voxel idx)
        // Non-short-circuit predicates keep this branchless (v_cmp + v_cndmask),
        // so EXEC stays all-ones throughout the WMMA region.
        #pragma unroll
        for (int off = 1; off < 16; off <<= 1) {
            #pragma unroll
            for (int r = 0; r < 8; ++r) {
                const float s2 = __shfl_xor(best[r], off, 32);
                const int   i2 = __shfl_xor(bidx[r], off, 32);
                const bool take = (s2 < best[r]) |
                                  ((s2 == best[r]) & (i2 < bidx[r]));
                best[r] = take ? s2 : best[r];
                bidx[r] = take ? i2 : bidx[r];
            }
        }
        if (M == 0) {
            #pragma unroll
            for (int r = 0; r < 8; ++r)
                sel[rt * 16 + r + (hiHalf ? 8 : 0)] = bidx[r];
        }
    }
    __syncthreads();

    // ---- Phase 3: stable rank = count of earlier points with same sel ----
    for (int n = tid; n < N_PTS; n += 256) {
        const int sv = sel[n];
        int r = 0;
        for (int n2 = 0; n2 < n; ++n2) r += (sel[n2] == sv);
        rnk[n] = r;
    }
    __syncthreads();

    // ---- Phase 4: scatter rows (feat ++ xyz  ==  input row rotated by 3) ----
    for (int n = wv; n < N_PTS; n += 8) {
        const int r = rnk[n];
        if (r < MAX_PTS) {
            const int vx = sel[n];
            const size_t ob = (((size_t)p * MAX_VOX + vx) * MAX_PTS + r) * ROWLEN;
            const size_t ib = ((size_t)p * N_PTS + n) * ROWLEN;
            for (int j = lane; j < ROWLEN; j += 32)
                out[ob + j] = xf[ib + ((j + 3) & 127)];
        }
    }
}

// ---------------------------------------------------------------------------
extern "C" void kernel_launch(void* const* d_in, const int* in_sizes, int n_in,
                              void* d_out, int out_size, void* d_ws, size_t ws_size,
                              hipStream_t stream) {
    (void)in_sizes; (void)n_in; (void)d_ws; (void)ws_size;
    const float* prop = (const float*)d_in[0];   // (64, 7) f32
    const float* xf   = (const float*)d_in[1];   // (64, 512, 128) f32
    float* out = (float*)d_out;                  // (64, 1728, 4, 128) f32

    const long long n4 = (long long)out_size / 4;       // float4 count
    voxz_zero_kernel<<<8192, 256, 0, stream>>>((float4*)d_out, n4);
    voxelize_kernel<<<P_CNT, 256, 0, stream>>>(prop, xf, out);
}